// CosSim_62302795596397
// MI455X (gfx1250) — compile-verified
//
#include <hip/hip_runtime.h>
#include <math.h>

#define NUM_CLASSES 100
#define D 512
#define EPS 1e-8f

typedef __attribute__((ext_vector_type(2))) float v2f;
typedef __attribute__((ext_vector_type(8))) float v8f;

// -------- workspace layout (floats) --------
// [0, 51200)       : class sums  -> overwritten in place with class means
// [51200, 51300)   : class counts (float)
// [51300, 51400)   : 1 / max(||mean_c||, EPS)

__global__ __launch_bounds__(256) void zero_ws_kernel(float* ws, int n) {
    int i = blockIdx.x * blockDim.x + threadIdx.x;
    if (i < n) ws[i] = 0.f;
}

// -------- segment sum with LDS staging --------
#define COLS_PER_CHUNK 128
#define ACC_THREADS 256

__global__ __launch_bounds__(ACC_THREADS)
void accum_kernel(const float* __restrict__ feat, const int* __restrict__ label,
                  float* __restrict__ sums, float* __restrict__ counts,
                  int n, int rowsPerChunk) {
    __shared__ float lsum[NUM_CLASSES * COLS_PER_CHUNK];   // 51.2 KB
    __shared__ float lcnt[NUM_CLASSES];

    const int tid = threadIdx.x;
    for (int i = tid; i < NUM_CLASSES * COLS_PER_CHUNK; i += ACC_THREADS) lsum[i] = 0.f;
    if (blockIdx.y == 0)
        for (int i = tid; i < NUM_CLASSES; i += ACC_THREADS) lcnt[i] = 0.f;
    __syncthreads();

    const int col     = tid & (COLS_PER_CHUNK - 1);
    const int rl      = tid >> 7;                 // 0..1: two rows in flight
    const int colBase = blockIdx.y * COLS_PER_CHUNK;
    const int r0      = blockIdx.x * rowsPerChunk;
    const int r1      = min(r0 + rowsPerChunk, n);

    for (int r = r0 + rl; r < r1; r += 2) {
        int   c = label[r];
        float v = feat[(long long)r * D + colBase + col];
        atomicAdd(&lsum[c * COLS_PER_CHUNK + col], v);      // ds_add_f32
        if (blockIdx.y == 0 && col == 0) atomicAdd(&lcnt[c], 1.f);
    }
    __syncthreads();

    for (int i = tid; i < NUM_CLASSES * COLS_PER_CHUNK; i += ACC_THREADS) {
        int   c  = i >> 7;
        int   cc = i & (COLS_PER_CHUNK - 1);
        float v  = lsum[i];
        if (v != 0.f) atomicAdd(&sums[c * D + colBase + cc], v);
    }
    if (blockIdx.y == 0)
        for (int i = tid; i < NUM_CLASSES; i += ACC_THREADS) {
            float v = lcnt[i];
            if (v != 0.f) atomicAdd(&counts[i], v);
        }
}

// -------- means + per-class 1/max(norm, eps) --------
__global__ __launch_bounds__(128)
void finalize_kernel(float* __restrict__ means /* in: sums */,
                     const float* __restrict__ counts,
                     float* __restrict__ rcpn) {
    const int c   = blockIdx.x;
    const int tid = threadIdx.x;
    const float rc = 1.f / fmaxf(counts[c], 1.f);

    float p = 0.f;
    for (int j = tid; j < D; j += 128) {
        float m = means[c * D + j] * rc;
        means[c * D + j] = m;
        p += m * m;
    }
    __shared__ float red[128];
    red[tid] = p;
    __syncthreads();
    for (int s = 64; s > 0; s >>= 1) {
        if (tid < s) red[tid] += red[tid + s];
        __syncthreads();
    }
    if (tid == 0) rcpn[c] = 1.f / fmaxf(sqrtf(red[0]), EPS);
}

// -------- WMMA cosine similarity --------
__device__ __forceinline__ float vget8(v8f v, int i) {
    switch (i) {
        case 0: return v[0]; case 1: return v[1];
        case 2: return v[2]; case 3: return v[3];
        case 4: return v[4]; case 5: return v[5];
        case 6: return v[6]; default: return v[7];
    }
}

__global__ __launch_bounds__(256)
void cossim_kernel(const float* __restrict__ feat, const float* __restrict__ means,
                   const int* __restrict__ label, const float* __restrict__ rcpn,
                   float* __restrict__ out, int n) {
    const int lane = threadIdx.x & 31;
    const int wave = threadIdx.x >> 5;
    const long long base = ((long long)blockIdx.x * 8 + wave) * 16;
    if (base >= n) return;                    // wave-uniform

    if (base + 16 <= n) {
        // ---- full 16-sample tile: V_WMMA_F32_16X16X4_F32 path ----
        const int m  = lane & 15;             // matrix row M / B column N
        const int kh = (lane >> 4) * 2;       // K-half: lanes 0-15 -> K%4 in {0,1}; 16-31 -> {2,3}
        const int cls = label[base + m];

        const float* ap = feat  + (base + m) * (long long)D + kh;
        const float* bp = means + (long long)cls * D + kh;

        v8f  acc = {};
        float pf = 0.f;                       // partial ||feat||^2 (this lane's K slices)
        #pragma unroll 4
        for (int k = 0; k < D; k += 4) {
            float2 av = *reinterpret_cast<const float2*>(ap + k);
            float2 bv = *reinterpret_cast<const float2*>(bp + k);
            pf += av.x * av.x + av.y * av.y;
            v2f a = { av.x, av.y };
            v2f b = { bv.x, bv.y };
            acc = __builtin_amdgcn_wmma_f32_16x16x4_f32(
                      false, a, false, b, (short)0, acc, false, false);
        }

        // diagonal of C: VGPR r -> (M=r, lane=r) and (M=r+8, lane=r+24)
        float dval = 0.f;
        if (lane < 8)       dval = vget8(acc, lane);
        else if (lane >= 24) dval = vget8(acc, lane - 24);

        // sample j's diagonal element sits on lane j (j<8) or lane j+16 (j>=8)
        const int j   = lane & 15;
        const int src = (j < 8) ? j : (j + 16);
        float dot = __shfl(dval, src, 32);

        // full ||feat_j||^2 = lane j part + lane j+16 part
        float nf2 = pf + __shfl_xor(pf, 16, 32);

        if (lane < 16) {
            float nf = fmaxf(sqrtf(nf2), EPS);
            out[base + lane] = dot / nf * rcpn[cls];
        }
    } else {
        // ---- scalar tail (not hit for N=100000, kept for generality) ----
        const int rem = (int)(n - base);
        if (lane < rem) {
            long long r = base + lane;
            int cls = label[r];
            const float* fr = feat + r * (long long)D;
            const float* mr = means + (long long)cls * D;
            float dot = 0.f, nf2 = 0.f;
            for (int k = 0; k < D; ++k) {
                float fv = fr[k];
                dot += fv * mr[k];
                nf2 += fv * fv;
            }
            out[r] = dot / fmaxf(sqrtf(nf2), EPS) * rcpn[cls];
        }
    }
}

extern "C" void kernel_launch(void* const* d_in, const int* in_sizes, int n_in,
                              void* d_out, int out_size, void* d_ws, size_t ws_size,
                              hipStream_t stream) {
    const float* feat  = (const float*)d_in[0];
    const int*   label = (const int*)d_in[1];
    const int    n     = in_sizes[1];            // 100000 samples

    float* ws     = (float*)d_ws;
    float* sums   = ws;                          // NUM_CLASSES * D
    float* counts = ws + NUM_CLASSES * D;        // NUM_CLASSES
    float* rcpn   = counts + NUM_CLASSES;        // NUM_CLASSES

    const int zn = NUM_CLASSES * D + 2 * NUM_CLASSES;
    zero_ws_kernel<<<(zn + 255) / 256, 256, 0, stream>>>(ws, zn);

    const int ROW_CHUNKS = 64;
    const int rowsPerChunk = (n + ROW_CHUNKS - 1) / ROW_CHUNKS;
    dim3 ag(ROW_CHUNKS, D / COLS_PER_CHUNK);     // 64 x 4 blocks
    accum_kernel<<<ag, ACC_THREADS, 0, stream>>>(feat, label, sums, counts, n, rowsPerChunk);

    finalize_kernel<<<NUM_CLASSES, 128, 0, stream>>>(sums, counts, rcpn);

    const int nTiles  = (n + 15) / 16;
    const int nBlocks = (nTiles + 7) / 8;        // 8 waves/block, 16 samples/wave
    cossim_kernel<<<nBlocks, 256, 0, stream>>>(feat, sums, label, rcpn,
                                               (float*)d_out, n);
}